// GaussianMeanShift_40183714021769
// MI455X (gfx1250) — compile-verified
//
#include <hip/hip_runtime.h>

// ---------------------------------------------------------------------------
// Gaussian mean-shift on MI455X (gfx1250, wave32).
// Hot loop = hill-climb: fused  W = exp(-0.5*sqdist(Z,X)); num = W@[X|1]
// on the fp32 matrix pipe (V_WMMA_F32_16X16X4_F32), with X tiles staged in
// LDS via double-buffered GLOBAL_LOAD_ASYNC_TO_LDS_B128 (ASYNCcnt) when the
// toolchain exposes the builtins.  All reductions fixed-order/deterministic.
// ---------------------------------------------------------------------------

#define N_POINTS   262144
#define DIMS       64
#define NSEEDS     100
#define MPAD       128          // 8 seed tiles of 16 (rows >=100 are zero)
#define SEED_TILES 8
#define KE         68           // extended K: [ -2Z | z2 | 1 | 0 0 ]
#define KSTEPS     17           // KE / 4
#define XROW       80           // LDS row stride: 5 dim-tiles of 16
#define DTILES     5            // dims 0..63 plus [1, x2, 0..] tile (col 64 = row-sum)
#define NBLK       256          // blocks in hill kernel
#define PT_TILES   (N_POINTS / 16)
#define TILE_ITERS (PT_TILES / NBLK)   // 64, exact -> uniform pipeline
#define HTHREADS   256          // 8 waves per block
#define HILL_ITERS 10
#define NUM_CC_    20
#define EPS2       0.0025f      // EPSILON^2, EPSILON = 0.05
#define RB         1024         // reduction blocks for seed stage (1024*256 = N)
#define RT         256

typedef __attribute__((ext_vector_type(2))) float v2f;
typedef __attribute__((ext_vector_type(8))) float v8f;
typedef int v4i __attribute__((vector_size(16)));
typedef __attribute__((address_space(1))) v4i* gptr_v4i;  // global 128b chunk
typedef __attribute__((address_space(3))) v4i* lptr_v4i;  // LDS 128b chunk

// Only meaningful in the device pass; host pass doesn't declare amdgcn builtins.
#if defined(__HIP_DEVICE_COMPILE__) && \
    !__has_builtin(__builtin_amdgcn_wmma_f32_16x16x4_f32)
#error "gfx1250 device pass: __builtin_amdgcn_wmma_f32_16x16x4_f32 not available"
#endif

// async global->LDS path (CDNA5 GLOBAL_LOAD_ASYNC_TO_LDS_B128 + s_wait_asynccnt)
#if defined(__HIP_DEVICE_COMPILE__) &&                              \
    __has_builtin(__builtin_amdgcn_global_load_async_to_lds_b128) && \
    __has_builtin(__builtin_amdgcn_s_wait_asynccnt)
#define USE_ASYNC_LDS 1
#else
#define USE_ASYNC_LDS 0
#endif

// D = A(16x4 f32) * B(4x16 f32) + C(16x16 f32), wave32.
__device__ inline v8f wmma4(v2f a, v2f b, v8f c) {
  return __builtin_amdgcn_wmma_f32_16x16x4_f32(false, a, false, b, (short)0, c,
                                               false, false);
}

__device__ inline unsigned long long splitmix64(unsigned long long x) {
  x += 0x9E3779B97F4A7C15ull;
  x = (x ^ (x >> 30)) * 0xBF58476D1CE4E5B9ull;
  x = (x ^ (x >> 27)) * 0x94D049BB133111EBull;
  return x ^ (x >> 31);
}

// ---------------------------------------------------------------------------
// small utility: zero a float buffer
// ---------------------------------------------------------------------------
__global__ void zerobuf_kernel(float* __restrict__ p, int n) {
  int i = blockIdx.x * blockDim.x + threadIdx.x;
  if (i < n) p[i] = 0.0f;
}

// ---------------------------------------------------------------------------
// Seed selection (k-means++ style, deterministic splitmix64 sampler).
// ---------------------------------------------------------------------------
__global__ __launch_bounds__(RT) void seed_init_kernel(
    const float* __restrict__ X, float* __restrict__ Zbuf,
    float* __restrict__ minD, float* __restrict__ blkSums) {
  __shared__ float red[RT];
  const int j = blockIdx.x * RT + threadIdx.x;
  const unsigned idx0 = (unsigned)(splitmix64(0x0C0FFEEull) % N_POINTS);
  const float* s0 = X + (size_t)idx0 * DIMS;
  const float* xj = X + (size_t)j * DIMS;
  float d2 = 0.0f;
#pragma unroll
  for (int k = 0; k < DIMS; k += 4) {
    float4 a = *(const float4*)(xj + k);
    float4 b = *(const float4*)(s0 + k);
    float dx = a.x - b.x, dy = a.y - b.y, dz = a.z - b.z, dw = a.w - b.w;
    d2 += dx * dx + dy * dy + dz * dz + dw * dw;
  }
  const float dist = sqrtf(d2);
  minD[j] = dist;
  red[threadIdx.x] = dist;
  __syncthreads();
  for (int s = RT / 2; s > 0; s >>= 1) {
    if (threadIdx.x < s) red[threadIdx.x] += red[threadIdx.x + s];
    __syncthreads();
  }
  if (threadIdx.x == 0) blkSums[blockIdx.x] = red[0];
  if (blockIdx.x == 0 && threadIdx.x < DIMS)
    Zbuf[threadIdx.x] = s0[threadIdx.x];
}

// pick index with prob ~ minD (weighted sampling), copy X[idx] -> Zbuf[step]
__global__ __launch_bounds__(256) void seed_select_kernel(
    const float* __restrict__ X, const float* __restrict__ minD,
    const float* __restrict__ blkSums, float* __restrict__ Zbuf, int step) {
  __shared__ float red[256];
  __shared__ int pick;
  float t = 0.0f;
  for (int i = threadIdx.x; i < RB; i += 256) t += blkSums[i];
  red[threadIdx.x] = t;
  __syncthreads();
  for (int s = 128; s > 0; s >>= 1) {
    if (threadIdx.x < s) red[threadIdx.x] += red[threadIdx.x + s];
    __syncthreads();
  }
  if (threadIdx.x == 0) {
    const float total = red[0];
    unsigned long long h =
        splitmix64(0x5EEDull * 1315423911ull + (unsigned long long)step);
    const float u =
        (float)((double)(h >> 11) * (1.0 / 9007199254740992.0)) * total;
    float acc = 0.0f;
    int b = RB - 1;
    for (int i = 0; i < RB; ++i) {
      const float v = blkSums[i];
      if (acc + v >= u) { b = i; break; }
      acc += v;
    }
    const int base = b * RT;
    int idx = base + RT - 1;
    for (int e = 0; e < RT; ++e) {
      const float v = minD[base + e];
      if (acc + v >= u) { idx = base + e; break; }
      acc += v;
    }
    pick = idx;
  }
  __syncthreads();
  if (threadIdx.x < DIMS)
    Zbuf[(size_t)step * DIMS + threadIdx.x] =
        X[(size_t)pick * DIMS + threadIdx.x];
}

// minD = min(minD, ||X - Z[step]||), and per-block sums for next selection
__global__ __launch_bounds__(RT) void seed_update_kernel(
    const float* __restrict__ X, const float* __restrict__ Zbuf,
    float* __restrict__ minD, float* __restrict__ blkSums, int step) {
  __shared__ float red[RT];
  const int j = blockIdx.x * RT + threadIdx.x;
  const float* zs = Zbuf + (size_t)step * DIMS;
  const float* xj = X + (size_t)j * DIMS;
  float d2 = 0.0f;
#pragma unroll
  for (int k = 0; k < DIMS; k += 4) {
    float4 a = *(const float4*)(xj + k);
    float4 b = *(const float4*)(zs + k);
    float dx = a.x - b.x, dy = a.y - b.y, dz = a.z - b.z, dw = a.w - b.w;
    d2 += dx * dx + dy * dy + dz * dz + dw * dw;
  }
  const float nd = fminf(minD[j], sqrtf(d2));
  minD[j] = nd;
  red[threadIdx.x] = nd;
  __syncthreads();
  for (int s = RT / 2; s > 0; s >>= 1) {
    if (threadIdx.x < s) red[threadIdx.x] += red[threadIdx.x + s];
    __syncthreads();
  }
  if (threadIdx.x == 0) blkSums[blockIdx.x] = red[0];
}

// ---------------------------------------------------------------------------
// Build extended seed matrix Ze[MPAD][KE]: row s = [-2*Z[s], |Z[s]|^2, 1, 0, 0]
// (rows >= NSEEDS all-zero).  Then S' = Ze . [X|1|x2|0|0] == sqdist(Z, X).
// ---------------------------------------------------------------------------
__global__ void buildze_kernel(const float* __restrict__ Zbuf,
                               float* __restrict__ Ze) {
  int i = blockIdx.x * blockDim.x + threadIdx.x;
  if (i >= MPAD * KE) return;
  const int s = i / KE, k = i - s * KE;
  float v = 0.0f;
  if (s < NSEEDS) {
    if (k < DIMS) {
      v = -2.0f * Zbuf[s * DIMS + k];
    } else if (k == DIMS) {
      float z2 = 0.0f;
      for (int d = 0; d < DIMS; ++d) {
        const float z = Zbuf[s * DIMS + d];
        z2 += z * z;
      }
      v = z2;
    } else if (k == DIMS + 1) {
      v = 1.0f;
    }
  }
  Ze[i] = v;
}

// ---------------------------------------------------------------------------
// Hill-climb main kernel: 8 waves/block, wave w owns seed tile w (16 seeds).
// Per 16-point tile:
//   Stage A (17x WMMA f32 16x16x4): S[16s x 16p] = sqdist via Ze extension
//   W = exp(-0.5 * max(S,0)); transpose through per-wave LDS
//   Stage B (20x WMMA): acc[16s x 80] += W @ [X | 1 | x2 | 0...]  (col 64 = den)
// X tiles double-buffered in LDS via async global->LDS copies (ASYNCcnt):
// issue tile t+1 into the idle buffer, then compute on tile t.
// ---------------------------------------------------------------------------
__global__ __launch_bounds__(HTHREADS) void hill_kernel(
    const float* __restrict__ X, const float* __restrict__ Ze,
    float* __restrict__ partials) {
  __shared__ float Xlds[2][16 * XROW];
  __shared__ float Wlds[SEED_TILES * 256];

  const int tid = threadIdx.x;
  const int wave = tid >> 5;
  const int lane = tid & 31;
  const int hoff = (lane >= 16) ? 2 : 0;  // K half-offset for this half-wave
  const int l15 = lane & 15;

  // zero the extension columns 64..79 of both buffers once
  // (async copies only ever write cols 0..63; 64/65 rewritten per tile)
  for (int i = tid; i < 2 * 16 * 16; i += HTHREADS) {
    const int b = i >> 8, r = (i >> 4) & 15, c = i & 15;
    Xlds[b][r * XROW + 64 + c] = 0.0f;
  }

  // A-matrix (Ze rows for this wave's 16 seeds) lives in registers: 17 pairs
  v2f Areg[KSTEPS];
  {
    const int row = wave * 16 + l15;
#pragma unroll
    for (int t = 0; t < KSTEPS; ++t) {
      const int k0 = 4 * t + hoff;
      Areg[t].x = Ze[row * KE + k0];
      Areg[t].y = Ze[row * KE + k0 + 1];
    }
  }

  v8f acc[DTILES];
#pragma unroll
  for (int d = 0; d < DTILES; ++d) acc[d] = (v8f){};

  float* wl = &Wlds[wave * 256];
  const size_t tileFloats = (size_t)16 * DIMS;  // floats per 16-point tile

#if USE_ASYNC_LDS
  // each of the 256 lanes copies one 16B chunk -> exactly 1 async instr/wave
  const int ldsOff = (tid >> 4) * XROW + (tid & 15) * 4;
  {
    const float* g = X + (size_t)blockIdx.x * tileFloats + (size_t)tid * 4;
    __builtin_amdgcn_global_load_async_to_lds_b128(
        (gptr_v4i)g, (lptr_v4i)&Xlds[0][ldsOff], 0, 0);
  }
#endif
  __syncthreads();

  int cur = 0;
  for (int it = 0; it < TILE_ITERS; ++it) {
    const int tile = blockIdx.x + it * NBLK;

#if USE_ASYNC_LDS
    // issue next tile into the idle buffer (its readers were released by the
    // trailing barrier of the previous iteration), then wait for current.
    if (it + 1 < TILE_ITERS) {
      const float* g =
          X + (size_t)(tile + NBLK) * tileFloats + (size_t)tid * 4;
      __builtin_amdgcn_global_load_async_to_lds_b128(
          (gptr_v4i)g, (lptr_v4i)&Xlds[cur ^ 1][ldsOff], 0, 0);
      __builtin_amdgcn_s_wait_asynccnt(1);  // older (cur) load done
    } else {
      __builtin_amdgcn_s_wait_asynccnt(0);
    }
    __syncthreads();
#else
    // synchronous fallback: cooperative load of the 16x64 tile
    {
      const float* xt = X + (size_t)tile * tileFloats;
      for (int i = tid; i < 16 * DIMS; i += HTHREADS) {
        const int p = i >> 6, k = i & 63;
        Xlds[cur][p * XROW + k] = xt[i];
      }
      if (it + 1 < TILE_ITERS && tid < 32)
        __builtin_prefetch(xt + (size_t)NBLK * tileFloats + tid * 32, 0, 0);
    }
    __syncthreads();
#endif

    float* Xc = Xlds[cur];
    if (tid < 16) {
      float s = 0.0f;
      for (int k = 0; k < DIMS; ++k) {
        const float v = Xc[tid * XROW + k];
        s += v * v;
      }
      Xc[tid * XROW + 64] = 1.0f;  // ones column -> row-sum (denominator)
      Xc[tid * XROW + 65] = s;     // |x|^2 for the sqdist extension
    }
    __syncthreads();

    // ---- Stage A: S = sqdist(Ztile, Xtile), K = 68 ----
    v8f S = (v8f){};
#pragma unroll
    for (int t = 0; t < KSTEPS; ++t) {
      v2f b;
      b.x = Xc[l15 * XROW + 4 * t + hoff];
      b.y = Xc[l15 * XROW + 4 * t + hoff + 1];
      S = wmma4(Areg[t], b, S);
    }

    // ---- W = exp(-0.5 * max(S,0)); write transposed via per-wave LDS ----
#pragma unroll
    for (int v = 0; v < 8; ++v) {
      const float w = __expf(-0.5f * fmaxf(S[v], 0.0f));
      wl[(v + (hoff ? 8 : 0)) * 16 + l15] = w;  // [seedRow][point]
    }

    // ---- Stage B: acc += W(16x16) @ Xe(16x80) ----
    v2f Aw[4];
#pragma unroll
    for (int tp = 0; tp < 4; ++tp) {
      const int p0 = 4 * tp + hoff;
      Aw[tp].x = wl[l15 * 16 + p0];
      Aw[tp].y = wl[l15 * 16 + p0 + 1];
    }
#pragma unroll
    for (int dt = 0; dt < DTILES; ++dt) {
      const int dcol = dt * 16 + l15;
#pragma unroll
      for (int tp = 0; tp < 4; ++tp) {
        v2f b;
        b.x = Xc[(4 * tp + hoff) * XROW + dcol];
        b.y = Xc[(4 * tp + hoff + 1) * XROW + dcol];
        acc[dt] = wmma4(Aw[tp], b, acc[dt]);
      }
    }
    __syncthreads();  // release buf[cur] for the copy issued next iteration
    cur ^= 1;
  }

  // write this block's partial [MPAD x XROW] (256 lanes * 40 vals = 10240)
  float* P = partials + (size_t)blockIdx.x * (MPAD * XROW);
#pragma unroll
  for (int dt = 0; dt < DTILES; ++dt)
#pragma unroll
    for (int v = 0; v < 8; ++v) {
      const int row = wave * 16 + v + (hoff ? 8 : 0);
      P[row * XROW + dt * 16 + l15] = acc[dt][v];
    }
}

// fixed-order reduction over NBLK partials
__global__ void reduce_kernel(const float* __restrict__ partials,
                              float* __restrict__ numRed) {
  const int i = blockIdx.x * blockDim.x + threadIdx.x;
  if (i >= MPAD * XROW) return;
  float s = 0.0f;
  for (int b = 0; b < NBLK; ++b) s += partials[(size_t)b * MPAD * XROW + i];
  numRed[i] = s;
}

// Z = num / den (den is the ones-column, col 64)
__global__ void zdivide_kernel(const float* __restrict__ numRed,
                               float* __restrict__ Zbuf) {
  const int i = blockIdx.x * blockDim.x + threadIdx.x;
  if (i >= NSEEDS * DIMS) return;
  const int s = i >> 6;
  const float den = numRed[s * XROW + 64];
  Zbuf[i] = numRed[s * XROW + (i & 63)] / fmaxf(den, 1e-20f);
}

// ---------------------------------------------------------------------------
// Serial connected-components (faithful to reference) + cluster means output.
// ---------------------------------------------------------------------------
__global__ __launch_bounds__(128) void cc_final_kernel(
    const float* __restrict__ Zbuf, float* __restrict__ out) {
  __shared__ float Zs[NSEEDS * DIMS];
  __shared__ int labels[NSEEDS];
  __shared__ unsigned char maskArr[NSEEDS];
  __shared__ int counts[NSEEDS];
  __shared__ int newLabel, doFlag, Kc;
  const int tid = threadIdx.x;

  for (int i = tid; i < NSEEDS * DIMS; i += 128) Zs[i] = Zbuf[i];
  if (tid < NSEEDS) labels[tid] = -1;
  if (tid == 0) Kc = 0;
  __syncthreads();

  for (int i = 0; i < NSEEDS; ++i) {
    if (tid < NSEEDS) {
      float d2 = 0.0f;
      for (int k = 0; k < DIMS; ++k) {
        const float d = Zs[i * DIMS + k] - Zs[tid * DIMS + k];
        d2 += d * d;
      }
      maskArr[tid] = (d2 <= EPS2) ? 1 : 0;
    }
    __syncthreads();
    if (tid < NSEEDS) {
      int c = 0;
      for (int j = 0; j < NSEEDS; ++j)
        c += (maskArr[j] && labels[j] == tid) ? 1 : 0;
      counts[tid] = c;
    }
    __syncthreads();
    if (tid == 0) {
      int best = 0, bl = 0;
      for (int l = 0; l < NSEEDS; ++l)
        if (counts[l] > best) { best = counts[l]; bl = l; }  // ties -> smallest
      const int has = best > 0;
      const int dof = (labels[i] == -1) ? 1 : 0;
      newLabel = has ? bl : Kc;
      if (dof && !has) Kc++;
      doFlag = dof;
    }
    __syncthreads();
    if (tid < NSEEDS && doFlag && maskArr[tid]) labels[tid] = newLabel;
    __syncthreads();
  }

  // out[c][d] = sum_{labels==c} Z[s][d] / (count_c + 1e-8), c < NUM_CC
  for (int idx = tid; idx < NUM_CC_ * DIMS; idx += 128) {
    const int c = idx >> 6, d = idx & 63;
    float s = 0.0f;
    int cnt = 0;
    for (int j = 0; j < NSEEDS; ++j)
      if (labels[j] == c) { s += Zs[j * DIMS + d]; cnt++; }
    out[idx] = s / ((float)cnt + 1e-8f);
  }
}

// ---------------------------------------------------------------------------
extern "C" void kernel_launch(void* const* d_in, const int* in_sizes, int n_in,
                              void* d_out, int out_size, void* d_ws,
                              size_t ws_size, hipStream_t stream) {
  (void)in_sizes; (void)n_in; (void)out_size; (void)ws_size;
  const float* X = (const float*)d_in[0];  // [262144, 64] fp32
  float* out = (float*)d_out;              // [20, 64] fp32
  float* ws = (float*)d_ws;

  float* minD = ws;                          // N_POINTS
  float* blkSums = minD + N_POINTS;          // RB
  float* Zbuf = blkSums + RB;                // MPAD*DIMS
  float* Ze = Zbuf + MPAD * DIMS;            // MPAD*KE
  float* numRed = Ze + MPAD * KE;            // MPAD*XROW
  float* partials = numRed + MPAD * XROW;    // NBLK*MPAD*XROW (~10.5 MB)

  // seed selection
  zerobuf_kernel<<<(MPAD * DIMS + 255) / 256, 256, 0, stream>>>(Zbuf,
                                                                MPAD * DIMS);
  seed_init_kernel<<<RB, RT, 0, stream>>>(X, Zbuf, minD, blkSums);
  for (int s = 1; s < NSEEDS; ++s) {
    seed_select_kernel<<<1, 256, 0, stream>>>(X, minD, blkSums, Zbuf, s);
    if (s < NSEEDS - 1)
      seed_update_kernel<<<RB, RT, 0, stream>>>(X, Zbuf, minD, blkSums, s);
  }

  // hill climb
  buildze_kernel<<<(MPAD * KE + 255) / 256, 256, 0, stream>>>(Zbuf, Ze);
  for (int it = 0; it < HILL_ITERS; ++it) {
    hill_kernel<<<NBLK, HTHREADS, 0, stream>>>(X, Ze, partials);
    reduce_kernel<<<(MPAD * XROW + 255) / 256, 256, 0, stream>>>(partials,
                                                                 numRed);
    zdivide_kernel<<<(NSEEDS * DIMS + 255) / 256, 256, 0, stream>>>(numRed,
                                                                    Zbuf);
    if (it < HILL_ITERS - 1)
      buildze_kernel<<<(MPAD * KE + 255) / 256, 256, 0, stream>>>(Zbuf, Ze);
  }

  // connected components + cluster means
  cc_final_kernel<<<1, 128, 0, stream>>>(Zbuf, out);
}